// SO3Convolution_27393301414329
// MI455X (gfx1250) — compile-verified
//
#include <hip/hip_runtime.h>
#include <hip/hip_bf16.h>
#include <math.h>

// ---------------- fixed problem geometry ----------------
// x: [8,16,64,64,64] f32; kernel: [16,32,24] f32; bias: [32] f32
// out: [8,32,32,32,32] f32
#define NB    8
#define FIN   16
#define FOUT  32
#define BIN2  64          // 2*B_IN
#define BOUT2 32          // 2*B_INV
#define LMAX  16          // B_OUT degrees l = 0..15
#define NFREQ 31          // centered freqs -15..15
#define NMN   961         // 31*31
#define NGRID 24
#define SCALING_C 0.14433756729740643f   // 1/sqrt(48)
#define PI_F 3.14159265358979323846f

typedef __attribute__((ext_vector_type(2))) float v2f;
typedef __attribute__((ext_vector_type(8))) float v8f;
typedef __attribute__((ext_vector_type(4))) unsigned int v4u;
typedef __attribute__((ext_vector_type(4))) int v4i;
typedef __attribute__((ext_vector_type(8))) int v8i;

#if defined(__has_builtin)
#if __has_builtin(__builtin_amdgcn_tensor_load_to_lds)
#define HAVE_TDM 1
#endif
#endif
#ifndef HAVE_TDM
#define HAVE_TDM 0
#endif

// fp32 WMMA: D(16x16) = A(16x4) * B(4x16) + C   -> v_wmma_f32_16x16x4_f32
__device__ __forceinline__ v8f wmma4(v2f a, v2f b, v8f c) {
  return __builtin_amdgcn_wmma_f32_16x16x4_f32(false, a, false, b, (short)0, c,
                                               false, false);
}

__device__ __forceinline__ unsigned ldsoff(const void* p) {
  return (unsigned)(size_t)p;  // generic LDS address: low 32 bits = LDS offset
}

// 1D contiguous DMA of nelem f32 from global to LDS via Tensor Data Mover.
// D# group0: count=1, lds_addr, global_addr, type=2. group1: data_size=4B,
// tensor_dim0=tile_dim0=nelem, tensor_dim1=1, stride0=nelem. Tracked by TENSORcnt.
__device__ __forceinline__ void tdm_load_1d(unsigned lds, const void* gptr,
                                            unsigned nelem) {
#if HAVE_TDM
  unsigned long long ga = (unsigned long long)(size_t)gptr;
  v4u g0;
  g0.x = 1u;                                            // count = 1 descriptor
  g0.y = lds;                                           // lds_addr
  g0.z = (unsigned)(ga & 0xffffffffu);                  // global_addr lo
  g0.w = (unsigned)((ga >> 32) & 0x1ffffffu) | (2u << 30);  // addr hi | type=2
  v8i g1;
  g1[0] = (int)(2u << 16);                              // data_size = 4 bytes
  g1[1] = (int)((nelem & 0xffffu) << 16);               // tensor_dim0[15:0]
  g1[2] = (int)(((nelem >> 16) & 0xffffu) | (1u << 16)); // dim0 hi | tensor_dim1=1
  g1[3] = (int)((nelem & 0xffffu) << 16);               // tile_dim0
  g1[4] = 0;                                            // tile_dim1 = 0 (1D)
  g1[5] = (int)nelem;                                   // tensor_dim0_stride lo
  g1[6] = 0;
  g1[7] = 0;
  v4i z4 = {0, 0, 0, 0};
#if __clang_major__ >= 23
  v8i z8 = {0, 0, 0, 0, 0, 0, 0, 0};
  __builtin_amdgcn_tensor_load_to_lds(g0, g1, z4, z4, z8, 0);
#else
  __builtin_amdgcn_tensor_load_to_lds(g0, g1, z4, z4, 0);
#endif
#else
  (void)lds; (void)gptr; (void)nelem;
#endif
}

// ---------------- Wigner-d on device ----------------
__device__ __forceinline__ float lnfac(int n) { return lgammaf((float)n + 1.0f); }

__device__ float wigner_d(int l, int mp, int mm, float beta) {
  float c = cosf(0.5f * beta), s = sinf(0.5f * beta);
  float pref = 0.5f * (lnfac(l + mp) + lnfac(l - mp) + lnfac(l + mm) + lnfac(l - mm));
  int t0 = max(0, mm - mp), t1 = min(l + mm, l - mp);
  float acc = 0.0f;
  for (int t = t0; t <= t1; ++t) {
    float lncoef = pref - lnfac(l + mm - t) - lnfac(t) - lnfac(mp - mm + t) - lnfac(l - mp - t);
    float coef = (((mp - mm + t) & 1) ? -1.0f : 1.0f) * expf(lncoef);
    acc += coef * powf(c, (float)(2 * l - 2 * t + mm - mp)) * powf(s, (float)(mp - mm + 2 * t));
  }
  return acc;
}

__device__ float quad_w(int j) {  // SOFT weights, b = 32
  float sum = 0.0f;
  for (int k = 0; k < 32; ++k)
    sum += sinf((float)((2 * j + 1) * (2 * k + 1)) * PI_F / 128.0f) / (float)(2 * k + 1);
  float w = (2.0f / 32.0f) * sinf((float)(2 * j + 1) * PI_F / 128.0f) * sum;
  return w / 8192.0f;  // / (2*(2b)^2)
}

// ---------------- table kernels (zero-padded 31x31 grid) ----------------
__global__ __launch_bounds__(256) void k_tab_dfwd(float* __restrict__ dfwd) {
  int idx = blockIdx.x * 256 + threadIdx.x;          // [l][k=64][mn]
  if (idx >= LMAX * BIN2 * NMN) return;
  int mn = idx % NMN, k = (idx / NMN) % BIN2, l = idx / (NMN * BIN2);
  int m = mn / NFREQ - 15, n = mn % NFREQ - 15;
  float v = 0.0f;
  if (abs(m) <= l && abs(n) <= l) {
    float beta = PI_F * (float)(2 * k + 1) / 128.0f;
    v = wigner_d(l, m, n, beta) * quad_w(k);
  }
  dfwd[idx] = v;
}

__global__ __launch_bounds__(256) void k_tab_dinv(float* __restrict__ dinv) {
  int idx = blockIdx.x * 256 + threadIdx.x;          // [l][k=32][mn]
  if (idx >= LMAX * BOUT2 * NMN) return;
  int mn = idx % NMN, k = (idx / NMN) % BOUT2, l = idx / (NMN * BOUT2);
  int m = mn / NFREQ - 15, n = mn % NFREQ - 15;
  float v = 0.0f;
  if (abs(m) <= l && abs(n) <= l) {
    float beta = PI_F * (float)(2 * k + 1) / 64.0f;
    v = wigner_d(l, m, n, beta) * (float)(2 * l + 1);
  }
  dinv[idx] = v;
}

__global__ __launch_bounds__(256) void k_tab_frft(float* __restrict__ frre,
                                                  float* __restrict__ frim) {
  int idx = blockIdx.x * 256 + threadIdx.x;          // [l][g=24][mn]
  if (idx >= LMAX * NGRID * NMN) return;
  int mn = idx % NMN, g = (idx / NMN) % NGRID, l = idx / (NMN * NGRID);
  int m = mn / NFREQ - 15, n = mn % NFREQ - 15;
  float re = 0.0f, im = 0.0f;
  if (abs(m) <= l && abs(n) <= l) {
    int a = g >> 2, bi = (g >> 1) & 1, gi = g & 1;
    float alpha = (float)a * (PI_F / 3.0f);
    float beta  = (float)(bi + 1) * (PI_F / 16.0f);
    float gamma = (float)gi * PI_F;
    float d = wigner_d(l, m, n, beta);
    float ph = (float)m * alpha + (float)n * gamma;   // conj(D) = e^{+i(ma+ng)} d
    re = d * cosf(ph);
    im = d * sinf(ph);
  }
  frre[idx] = re;
  frim[idx] = im;
}

__global__ __launch_bounds__(256) void k_tab_tw(float* __restrict__ e64re,
                                                float* __restrict__ e64im,
                                                float* __restrict__ w32re,
                                                float* __restrict__ w32im) {
  int i = blockIdx.x * 256 + threadIdx.x;
  if (i < NFREQ * BIN2) {                            // E64[mi][a] = e^{-2pi i (mi-15) a / 64}
    int mi = i / BIN2, a = i % BIN2;
    float ang = -2.0f * PI_F * (float)(mi - 15) * (float)a / 64.0f;
    e64re[i] = cosf(ang);
    e64im[i] = sinf(ang);
  } else if (i < NFREQ * BIN2 + BOUT2 * NFREQ) {     // W32[a][mi] = e^{+2pi i (mi-15) a / 32}
    int j = i - NFREQ * BIN2;
    int a = j / NFREQ, mi = j % NFREQ;
    float ang = 2.0f * PI_F * (float)(mi - 15) * (float)a / 32.0f;
    w32re[j] = cosf(ang);
    w32im[j] = sinf(ang);
  }
}

// ------------- stage B (WMMA + TDM): pruned forward DFT per (b,f,beta) slice
// Xf[mn][k][b*16+f] = sum_{a,g} x[a,g] e^{-2pi i(ma+ng)/64}, m,n in -15..15
// step1: T = E(32x64,cplx) * X(64x64,real)     step2: Xf = T * E2^T (cplx*cplx)
__global__ __launch_bounds__(256) void k_fwd(const float* __restrict__ x,
                                             const float* __restrict__ e64re,
                                             const float* __restrict__ e64im,
                                             float* __restrict__ xfre,
                                             float* __restrict__ xfim) {
  __shared__ float xs[BIN2 * BIN2];     // x slice [a][g]
  __shared__ float ert[BIN2 * 32];      // E^T [a][mi], mi=31 zero-padded
  __shared__ float eit[BIN2 * 32];
  __shared__ float ttr[BIN2 * 32];      // T^T [g][mi]
  __shared__ float tti[BIN2 * 32];
  int sid = blockIdx.x;                 // b*1024 + f*64 + k
  const float* xp = x + (size_t)sid * (BIN2 * BIN2);
#if HAVE_TDM
  if (threadIdx.x < 32) tdm_load_1d(ldsoff(xs), xp, BIN2 * BIN2);
  __builtin_amdgcn_s_wait_tensorcnt(0);
#else
  for (int i = threadIdx.x; i < BIN2 * BIN2; i += 256) xs[i] = xp[i];
#endif
  for (int i = threadIdx.x; i < BIN2 * 32; i += 256) {   // transpose + pad
    int a = i >> 5, mi = i & 31;
    ert[i] = (mi < NFREQ) ? e64re[mi * BIN2 + a] : 0.0f;
    eit[i] = (mi < NFREQ) ? e64im[mi * BIN2 + a] : 0.0f;
  }
  __syncthreads();

  int lane = threadIdx.x & 31, wave = threadIdx.x >> 5, khalf = lane >> 4;
  // step1: 8 tiles (mt 0..1 x nt 0..3), one per wave; M=mi(32), N=g(64), K=a(64)
  {
    int mt = wave >> 2, nt = wave & 3;
    int row = mt * 16 + (lane & 15);
    int col = nt * 16 + (lane & 15);
    v8f cr = {0.f, 0.f, 0.f, 0.f, 0.f, 0.f, 0.f, 0.f};
    v8f ci = cr;
    for (int k0 = 0; k0 < BIN2; k0 += 4) {
      int ka = k0 + 2 * khalf;
      v2f ar, ai, b;
      ar.x = ert[ka * 32 + row];       ar.y = ert[(ka + 1) * 32 + row];
      ai.x = eit[ka * 32 + row];       ai.y = eit[(ka + 1) * 32 + row];
      b.x  = xs[ka * BIN2 + col];      b.y  = xs[(ka + 1) * BIN2 + col];
      cr = wmma4(ar, b, cr);
      ci = wmma4(ai, b, ci);
    }
#pragma unroll
    for (int j = 0; j < 8; ++j) {
      int mi = mt * 16 + j + 8 * khalf;
      ttr[col * 32 + mi] = cr[j];      // store transposed T^T[g][mi]
      tti[col * 32 + mi] = ci[j];
    }
  }
  __syncthreads();

  // step2: 4 tiles (mt 0..1 x nt 0..1) on waves 0..3; M=mi, N=ni, K=g(64), cplx
  if (wave < 4) {
    int mt = wave >> 1, nt = wave & 1;
    int row = mt * 16 + (lane & 15);
    int col = nt * 16 + (lane & 15);
    v8f cr = {0.f, 0.f, 0.f, 0.f, 0.f, 0.f, 0.f, 0.f};
    v8f ci = cr;
    for (int k0 = 0; k0 < BIN2; k0 += 4) {
      int ka = k0 + 2 * khalf;
      v2f ar, ai, ain, br, bi;
      ar.x = ttr[ka * 32 + row];       ar.y = ttr[(ka + 1) * 32 + row];
      ai.x = tti[ka * 32 + row];       ai.y = tti[(ka + 1) * 32 + row];
      br.x = ert[ka * 32 + col];       br.y = ert[(ka + 1) * 32 + col];
      bi.x = eit[ka * 32 + col];       bi.y = eit[(ka + 1) * 32 + col];
      ain.x = -ai.x;                   ain.y = -ai.y;
      cr = wmma4(ar, br, cr);          // Re: Tre*Ere - Tim*Eim
      cr = wmma4(ain, bi, cr);
      ci = wmma4(ar, bi, ci);          // Im: Tre*Eim + Tim*Ere
      ci = wmma4(ai, br, ci);
    }
    int b = sid >> 10, f = (sid >> 6) & 15, k = sid & 63;
#pragma unroll
    for (int j = 0; j < 8; ++j) {
      int mi = mt * 16 + j + 8 * khalf;
      if (mi < NFREQ && col < NFREQ) {
        size_t o = ((size_t)(mi * NFREQ + col) * BIN2 + k) * 128 + b * 16 + f;
        xfre[o] = cr[j];
        xfim[o] = ci[j];
      }
    }
  }
}

// ------------- stage C (WMMA + TDM): beta contraction per (m,n) pair -------------
// XL[l][mn][b*16+f] = sum_k dfwd[l][k][mn] * Xf[mn][k][bf]
// GEMM  A[16 x 64] (real) x B[64 x 128] (re / im); B panels DMA'd to LDS.
__global__ __launch_bounds__(256) void k_xl(const float* __restrict__ dfwd,
                                            const float* __restrict__ xfre,
                                            const float* __restrict__ xfim,
                                            float* __restrict__ xlre,
                                            float* __restrict__ xlim) {
  __shared__ float bres[BIN2 * 128];    // 32 KB contiguous B panel (re)
  __shared__ float bims[BIN2 * 128];    // 32 KB (im)
  int mn = blockIdx.x;                  // 0..960
#if HAVE_TDM
  if (threadIdx.x < 32) {
    tdm_load_1d(ldsoff(bres), xfre + (size_t)mn * BIN2 * 128, BIN2 * 128);
    tdm_load_1d(ldsoff(bims), xfim + (size_t)mn * BIN2 * 128, BIN2 * 128);
  }
  __builtin_amdgcn_s_wait_tensorcnt(0);
#else
  for (int i = threadIdx.x; i < BIN2 * 128; i += 256) {
    bres[i] = xfre[(size_t)mn * BIN2 * 128 + i];
    bims[i] = xfim[(size_t)mn * BIN2 * 128 + i];
  }
#endif
  __syncthreads();
  int lane = threadIdx.x & 31, wave = threadIdx.x >> 5, khalf = lane >> 4;
  int row = lane & 15;                  // A row = l
  int col = wave * 16 + (lane & 15);    // N col = b*16+f
  v8f cre = {0.f, 0.f, 0.f, 0.f, 0.f, 0.f, 0.f, 0.f};
  v8f cim = cre;
  for (int k0 = 0; k0 < BIN2; k0 += 4) {
    int ka = k0 + 2 * khalf;
    v2f a, br, bi;
    a.x = dfwd[((size_t)row * BIN2 + ka) * NMN + mn];
    a.y = dfwd[((size_t)row * BIN2 + ka + 1) * NMN + mn];
    br.x = bres[ka * 128 + col];   br.y = bres[(ka + 1) * 128 + col];
    bi.x = bims[ka * 128 + col];   bi.y = bims[(ka + 1) * 128 + col];
    cre = wmma4(a, br, cre);
    cim = wmma4(a, bi, cim);
  }
#pragma unroll
  for (int j = 0; j < 8; ++j) {
    int l = j + 8 * khalf;
    size_t o = ((size_t)l * NMN + mn) * 128 + col;
    xlre[o] = cre[j];
    xlim[o] = cim[j];
  }
}

// ------------- stage D: kernel spectrum Yl[l][mn][f*32+o] -------------
__global__ __launch_bounds__(256) void k_yl(const float* __restrict__ kern,
                                            const float* __restrict__ frre,
                                            const float* __restrict__ frim,
                                            float* __restrict__ ylre,
                                            float* __restrict__ ylim) {
  size_t idx = (size_t)blockIdx.x * 256 + threadIdx.x;   // (l*961+mn)*512 + f*32+o
  if (idx >= (size_t)LMAX * NMN * 512) return;
  int fo = (int)(idx & 511);
  size_t lmn = idx >> 9;
  int l = (int)(lmn / NMN), mn = (int)(lmn % NMN);
  int m = mn / NFREQ - 15, n = mn % NFREQ - 15;
  float ar = 0.0f, ai = 0.0f;
  if (abs(m) <= l && abs(n) <= l) {
    const float* kp = kern + (size_t)fo * NGRID;   // kernel[f][o][g], fo = f*32+o
    for (int g = 0; g < NGRID; ++g) {
      float kv = kp[g] * SCALING_C;
      size_t fi = ((size_t)l * NGRID + g) * NMN + mn;
      ar = fmaf(kv, frre[fi], ar);
      ai = fmaf(kv, frim[fi], ai);
    }
  }
  ylre[idx] = ar;
  ylim[idx] = ai;
}

// ------------- stage E (WMMA): per-degree complex GEMM -------------
// Z[b,o,m,n] = sum_{k,f} X[b,m,k,f] * conj(Y)[n,k,f,o]
__global__ __launch_bounds__(256) void k_z(const float* __restrict__ xlre,
                                           const float* __restrict__ xlim,
                                           const float* __restrict__ ylre,
                                           const float* __restrict__ ylim,
                                           float* __restrict__ zlre,
                                           float* __restrict__ zlim) {
  int l = blockIdx.y;
  int Lw = 2 * l + 1;
  int Md = NB * Lw, Nd = FOUT * Lw, Kd = FIN * Lw;
  int mt = blockIdx.x >> 3;
  int nwg = blockIdx.x & 7;
  if (mt * 16 >= Md || nwg * 128 >= Nd) return;      // uniform per block
  int lane = threadIdx.x & 31;
  int wave = threadIdx.x >> 5;
  int khalf = lane >> 4;
  int r = mt * 16 + (lane & 15);
  int c = nwg * 128 + wave * 16 + (lane & 15);
  bool rok = r < Md, cok = c < Nd;
  int b = rok ? r / Lw : 0, mp = rok ? r % Lw : 0;
  int n = cok ? c / FOUT : 0, o = cok ? c % FOUT : 0;
  int mi = mp - l + 15;
  int ni = n - l + 15;
  size_t lbase = (size_t)l * NMN;
  __builtin_prefetch(ylre + (lbase + (size_t)ni * NFREQ) * 512, 0, 0);
  v8f cr = {0.f, 0.f, 0.f, 0.f, 0.f, 0.f, 0.f, 0.f};
  v8f ci = cr;
  for (int k0 = 0; k0 < Kd; k0 += 4) {
    v2f are, aim, anre, yre, yim;
#pragma unroll
    for (int h = 0; h < 2; ++h) {
      int kk = k0 + 2 * khalf + h;
      int n2 = kk >> 4, f = kk & 15;
      int ki = n2 - l + 15;
      float vr = 0.f, vi = 0.f, wr = 0.f, wi = 0.f;
      if (rok) {
        size_t ao = (lbase + (size_t)mi * NFREQ + ki) * 128 + b * 16 + f;
        vr = xlre[ao];
        vi = xlim[ao];
      }
      if (cok) {
        size_t bo = (lbase + (size_t)ni * NFREQ + ki) * 512 + f * 32 + o;
        wr = ylre[bo];
        wi = ylim[bo];
      }
      if (h == 0) { are.x = vr; aim.x = vi; yre.x = wr; yim.x = wi; }
      else        { are.y = vr; aim.y = vi; yre.y = wr; yim.y = wi; }
    }
    anre.x = -are.x;
    anre.y = -are.y;
    cr = wmma4(are, yre, cr);     // Re: +Are*Yre + Aim*Yim
    cr = wmma4(aim, yim, cr);
    ci = wmma4(aim, yre, ci);     // Im: +Aim*Yre - Are*Yim
    ci = wmma4(anre, yim, ci);
  }
#pragma unroll
  for (int j = 0; j < 8; ++j) {
    int rr = mt * 16 + j + 8 * khalf;
    if (rr < Md && cok) {
      int b2 = rr / Lw, mp2 = rr % Lw;
      size_t zo = (lbase + (size_t)(mp2 - l + 15) * NFREQ + ni) * 256 + b2 * 32 + o;
      zlre[zo] = cr[j];
      zlim[zo] = ci[j];
    }
  }
}

// ------------- stage F (WMMA): inverse-beta expansion, sum over l -------------
// G[(b*32+o)*32+k][mn] = sum_l Z[l][mn][bo] * dinv[l][k][mn]  (dinv zero-padded)
// GEMM A[256 x 16](cplx) x B[16 x 32](real) per (m,n); 32 tiles on 8 waves.
__global__ __launch_bounds__(256) void k_g(const float* __restrict__ zlre,
                                           const float* __restrict__ zlim,
                                           const float* __restrict__ dinv,
                                           float* __restrict__ gre,
                                           float* __restrict__ gim) {
  int mn = blockIdx.x;
  int lane = threadIdx.x & 31, wave = threadIdx.x >> 5, khalf = lane >> 4;
  for (int tt = 0; tt < 4; ++tt) {
    int tile = wave * 4 + tt;
    int mt = tile >> 1, nt = tile & 1;
    int r = mt * 16 + (lane & 15);        // bo row
    int col = nt * 16 + (lane & 15);      // k_out col
    v8f cre = {0.f, 0.f, 0.f, 0.f, 0.f, 0.f, 0.f, 0.f};
    v8f cim = cre;
    for (int k0 = 0; k0 < LMAX; k0 += 4) {
      int ka = k0 + 2 * khalf;            // l index
      size_t a0 = ((size_t)ka * NMN + mn) * 256 + r;
      size_t a1 = ((size_t)(ka + 1) * NMN + mn) * 256 + r;
      v2f ar, ai, b;
      ar.x = zlre[a0];  ar.y = zlre[a1];
      ai.x = zlim[a0];  ai.y = zlim[a1];
      b.x = dinv[((size_t)ka * BOUT2 + col) * NMN + mn];
      b.y = dinv[((size_t)(ka + 1) * BOUT2 + col) * NMN + mn];
      cre = wmma4(ar, b, cre);
      cim = wmma4(ai, b, cim);
    }
#pragma unroll
    for (int j = 0; j < 8; ++j) {
      int r2 = mt * 16 + j + 8 * khalf;
      size_t go = ((size_t)r2 * BOUT2 + col) * NMN + mn;
      gre[go] = cre[j];
      gim[go] = cim[j];
    }
  }
}

// ------------- stage G (WMMA): pruned 32x32 inverse DFT + bias -------------
// step1: T = W(32x32,cplx) * G(32x32,cplx)   step2: z = Re(T * W^T) + bias
__global__ __launch_bounds__(256) void k_idft(const float* __restrict__ gre,
                                              const float* __restrict__ gim,
                                              const float* __restrict__ w32re,
                                              const float* __restrict__ w32im,
                                              const float* __restrict__ bias,
                                              float* __restrict__ out) {
  __shared__ float gsr[1024], gsi[1024];   // G padded [mi][ni] 32x32
  __shared__ float wrt[1024], wit[1024];   // W^T [mi][a] 32x32 (pad row 31)
  __shared__ float t_r[1024], t_i[1024];   // T^T [ni][a]
  int sid = blockIdx.x;                    // (b*32+o)*32 + k
  int o = (sid >> 5) & 31;
  const float* gr = gre + (size_t)sid * NMN;
  const float* gi = gim + (size_t)sid * NMN;
  for (int i = threadIdx.x; i < 1024; i += 256) {
    int mi = i >> 5, ni = i & 31;
    bool ok = (mi < NFREQ) && (ni < NFREQ);
    gsr[i] = ok ? gr[mi * NFREQ + ni] : 0.0f;
    gsi[i] = ok ? gi[mi * NFREQ + ni] : 0.0f;
    int a = i >> 5, m2 = i & 31;           // wrt[m2][a] = W[a][m2]
    wrt[m2 * 32 + a] = (m2 < NFREQ) ? w32re[a * NFREQ + m2] : 0.0f;
    wit[m2 * 32 + a] = (m2 < NFREQ) ? w32im[a * NFREQ + m2] : 0.0f;
  }
  __syncthreads();
  int lane = threadIdx.x & 31, wave = threadIdx.x >> 5, khalf = lane >> 4;
  // step1 on waves 0..3: T[a][ni] = sum_mi W[a][mi] G[mi][ni]
  if (wave < 4) {
    int mt = wave >> 1, nt = wave & 1;
    int row = mt * 16 + (lane & 15);       // a
    int col = nt * 16 + (lane & 15);       // ni
    v8f cr = {0.f, 0.f, 0.f, 0.f, 0.f, 0.f, 0.f, 0.f};
    v8f ci = cr;
    for (int k0 = 0; k0 < 32; k0 += 4) {
      int ka = k0 + 2 * khalf;             // mi
      v2f ar, ai, ain, br, bi;
      ar.x = wrt[ka * 32 + row];  ar.y = wrt[(ka + 1) * 32 + row];
      ai.x = wit[ka * 32 + row];  ai.y = wit[(ka + 1) * 32 + row];
      br.x = gsr[ka * 32 + col];  br.y = gsr[(ka + 1) * 32 + col];
      bi.x = gsi[ka * 32 + col];  bi.y = gsi[(ka + 1) * 32 + col];
      ain.x = -ai.x;              ain.y = -ai.y;
      cr = wmma4(ar, br, cr);
      cr = wmma4(ain, bi, cr);
      ci = wmma4(ar, bi, ci);
      ci = wmma4(ai, br, ci);
    }
#pragma unroll
    for (int j = 0; j < 8; ++j) {
      int a = mt * 16 + j + 8 * khalf;
      t_r[col * 32 + a] = cr[j];           // store transposed T^T[ni][a]
      t_i[col * 32 + a] = ci[j];
    }
  }
  __syncthreads();
  // step2 on waves 0..3: z[a][g] = Re( sum_ni T[a][ni] * W[g][ni] )
  if (wave < 4) {
    int mt = wave >> 1, nt = wave & 1;
    int row = mt * 16 + (lane & 15);       // a
    int col = nt * 16 + (lane & 15);       // g
    v8f cr = {0.f, 0.f, 0.f, 0.f, 0.f, 0.f, 0.f, 0.f};
    for (int k0 = 0; k0 < 32; k0 += 4) {
      int ka = k0 + 2 * khalf;             // ni
      v2f ar, ain, br, bi;
      ar.x = t_r[ka * 32 + row];   ar.y = t_r[(ka + 1) * 32 + row];
      ain.x = -t_i[ka * 32 + row]; ain.y = -t_i[(ka + 1) * 32 + row];
      br.x = wrt[ka * 32 + col];   br.y = wrt[(ka + 1) * 32 + col];
      bi.x = wit[ka * 32 + col];   bi.y = wit[(ka + 1) * 32 + col];
      cr = wmma4(ar, br, cr);              // Re: Tre*Wre - Tim*Wim
      cr = wmma4(ain, bi, cr);
    }
    float bv = bias[o];
#pragma unroll
    for (int j = 0; j < 8; ++j) {
      int a = mt * 16 + j + 8 * khalf;
      out[(size_t)sid * 1024 + a * 32 + col] = cr[j] + bv;
    }
  }
}

// ---------------- host launcher ----------------
extern "C" void kernel_launch(void* const* d_in, const int* in_sizes, int n_in,
                              void* d_out, int out_size, void* d_ws, size_t ws_size,
                              hipStream_t stream) {
  (void)in_sizes; (void)n_in; (void)out_size; (void)ws_size;
  const float* x    = (const float*)d_in[0];
  const float* kern = (const float*)d_in[1];
  const float* bias = (const float*)d_in[2];
  float* out = (float*)d_out;

  size_t off = 0;
  auto take = [&](size_t nfloats) {
    float* p = (float*)((char*)d_ws + off);
    off += ((nfloats * sizeof(float)) + 255) & ~(size_t)255;
    return p;
  };
  float* dfwd  = take((size_t)LMAX * BIN2 * NMN);
  float* dinv  = take((size_t)LMAX * BOUT2 * NMN);
  float* frre  = take((size_t)LMAX * NGRID * NMN);
  float* frim  = take((size_t)LMAX * NGRID * NMN);
  float* e64re = take((size_t)NFREQ * BIN2);
  float* e64im = take((size_t)NFREQ * BIN2);
  float* w32re = take((size_t)BOUT2 * NFREQ);
  float* w32im = take((size_t)BOUT2 * NFREQ);
  float* xfre  = take((size_t)NMN * BIN2 * 128);
  float* xfim  = take((size_t)NMN * BIN2 * 128);
  float* xlre  = take((size_t)LMAX * NMN * 128);
  float* xlim  = take((size_t)LMAX * NMN * 128);
  float* ylre  = take((size_t)LMAX * NMN * 512);
  float* ylim  = take((size_t)LMAX * NMN * 512);
  float* zlre  = take((size_t)LMAX * NMN * 256);
  float* zlim  = take((size_t)LMAX * NMN * 256);
  float* gre   = take((size_t)NB * FOUT * BOUT2 * NMN);
  float* gim   = take((size_t)NB * FOUT * BOUT2 * NMN);

  // tables
  k_tab_dfwd<<<(LMAX * BIN2 * NMN + 255) / 256, 256, 0, stream>>>(dfwd);
  k_tab_dinv<<<(LMAX * BOUT2 * NMN + 255) / 256, 256, 0, stream>>>(dinv);
  k_tab_frft<<<(LMAX * NGRID * NMN + 255) / 256, 256, 0, stream>>>(frre, frim);
  k_tab_tw<<<(NFREQ * BIN2 + BOUT2 * NFREQ + 255) / 256, 256, 0, stream>>>(
      e64re, e64im, w32re, w32im);

  // pipeline
  k_fwd<<<NB * FIN * BIN2, 256, 0, stream>>>(x, e64re, e64im, xfre, xfim);
  k_xl<<<NMN, 256, 0, stream>>>(dfwd, xfre, xfim, xlre, xlim);
  k_yl<<<(int)(((size_t)LMAX * NMN * 512 + 255) / 256), 256, 0, stream>>>(
      kern, frre, frim, ylre, ylim);
  k_z<<<dim3(128, LMAX, 1), 256, 0, stream>>>(xlre, xlim, ylre, ylim, zlre, zlim);
  k_g<<<NMN, 256, 0, stream>>>(zlre, zlim, dinv, gre, gim);
  k_idft<<<NB * FOUT * BOUT2, 256, 0, stream>>>(gre, gim, w32re, w32im, bias, out);
}